// MutualInformation_82411832476296
// MI455X (gfx1250) — compile-verified
//
#include <hip/hip_runtime.h>

typedef float v2f __attribute__((ext_vector_type(2)));
typedef float v8f __attribute__((ext_vector_type(8)));

#define B_     256
#define STEP_  32
#define DF_    256
#define DZ_    64
#define LOG2PI 1.8378770664093453f
#define LOGC   16.3649557f   // ln(256 * 50000)

static __device__ __forceinline__ v8f wmma_f32(v2f a, v2f b, v8f c) {
    // V_WMMA_F32_16X16X4_F32: D = A(16x4) x B(4x16) + C, full fp32
    return __builtin_amdgcn_wmma_f32_16x16x4_f32(
        /*neg_a=*/false, a, /*neg_b=*/false, b,
        /*c_mod=*/(short)0, c, /*reuse_a=*/false, /*reuse_b=*/false);
}

// ---------- per-distribution constants: sum_d (m^2 e^2 + 2 lv + log2pi) ----------
__global__ void __launch_bounds__(256) k_const_f(const float* __restrict__ fm,
                                                 const float* __restrict__ flv,
                                                 float* __restrict__ cf) {
    int j = blockIdx.x * blockDim.x + threadIdx.x;
    if (j >= B_) return;
    float acc = 0.f;
    const float* mp  = fm  + j * DF_;
    const float* lvp = flv + j * DF_;
    for (int d = 0; d < DF_; ++d) {
        float lv = lvp[d], m = mp[d];
        float e2 = __expf(-2.f * lv);
        acc += m * m * e2 + 2.f * lv + LOG2PI;
    }
    cf[j] = acc;
}

__global__ void __launch_bounds__(256) k_const_z(const float* __restrict__ zm,
                                                 const float* __restrict__ zlv,
                                                 float* __restrict__ cz) {
    int t = blockIdx.x * blockDim.x + threadIdx.x;
    if (t >= B_ * STEP_) return;
    const float* mp  = zm  + t * DZ_;
    const float* lvp = zlv + t * DZ_;
    float acc = 0.f;
    for (int d = 0; d < DZ_; ++d) {
        float lv = lvp[d], m = mp[d];
        float e2 = __expf(-2.f * lv);
        acc += m * m * e2 + 2.f * lv + LOG2PI;
    }
    cz[t] = acc;  // layout cz[j*STEP_ + s]
}

// ---------- sum_f[i][j] via fp32 WMMA: one wave per 16x16 tile ----------
__global__ void __launch_bounds__(32) k_sumf(const float* __restrict__ fm,
                                             const float* __restrict__ flv,
                                             const float* __restrict__ fs,
                                             const float* __restrict__ cf,
                                             float* __restrict__ sumf) {
    int tile = blockIdx.x;
    int it = tile >> 4, jt = tile & 15;
    int lane = threadIdx.x;
    int half = lane >> 4, lr = lane & 15;
    int koff = half * 2;

    int arow = it * 16 + lr;   // A: sample row (M = lr)
    int bcol = jt * 16 + lr;   // B: dist column (N = lr)
    const float* ap  = fs  + arow * DF_ + koff;
    const float* lvp = flv + bcol * DF_ + koff;
    const float* mp  = fm  + bcol * DF_ + koff;

    v8f acc = {0.f, 0.f, 0.f, 0.f, 0.f, 0.f, 0.f, 0.f};
    for (int k = 0; k < DF_; k += 4) {
        float2 sa = *reinterpret_cast<const float2*>(ap + k);
        float2 lv = *reinterpret_cast<const float2*>(lvp + k);
        float2 mu = *reinterpret_cast<const float2*>(mp + k);
        v2f a1; a1.x = sa.x * sa.x;          a1.y = sa.y * sa.y;
        v2f b1; b1.x = __expf(-2.f * lv.x);  b1.y = __expf(-2.f * lv.y);
        v2f a2; a2.x = sa.x;                 a2.y = sa.y;
        v2f b2; b2.x = -2.f * mu.x * b1.x;   b2.y = -2.f * mu.y * b1.y;
        acc = wmma_f32(a1, b1, acc);  // dot(s^2, e^2)
        acc = wmma_f32(a2, b2, acc);  // -2 dot(s, m e^2)
    }
    float cj = cf[bcol];
    #pragma unroll
    for (int v = 0; v < 8; ++v) {
        int row = it * 16 + v + half * 8;
        sumf[row * B_ + jt * 16 + lr] = -0.5f * (acc[v] + cj);
    }
}

// ---------- logq_f: one wave per row ----------
__global__ void __launch_bounds__(256) k_logqf(const float* __restrict__ sumf,
                                               float* __restrict__ tf) {
    int wave = threadIdx.x >> 5, lane = threadIdx.x & 31;
    int row = blockIdx.x * 8 + wave;
    const float* p = sumf + row * B_;
    float mx = -3.4e38f;
    for (int j = lane; j < B_; j += 32) mx = fmaxf(mx, p[j]);
    for (int m = 16; m >= 1; m >>= 1) mx = fmaxf(mx, __shfl_xor(mx, m, 32));
    float sm = 0.f;
    for (int j = lane; j < B_; j += 32) sm += __expf(p[j] - mx);
    for (int m = 16; m >= 1; m >>= 1) sm += __shfl_xor(sm, m, 32);
    if (lane == 0) {
        float logq = mx + __logf(sm) - LOGC;
        tf[row] = fmaxf(-logq, 0.f);
    }
}

// ---------- fused z: WMMA tile GEMM + logsumexp over j, per (step, i-tile) ----------
__global__ void __launch_bounds__(256) k_z(const float* __restrict__ zm,
                                           const float* __restrict__ zlv,
                                           const float* __restrict__ zs,
                                           const float* __restrict__ cz,
                                           const float* __restrict__ sumf,
                                           float* __restrict__ tz,
                                           float* __restrict__ tfz) {
    __shared__ float lz[16][B_ + 1];
    __shared__ float lfz[16][B_ + 1];
    int s  = blockIdx.x >> 4;   // 0..31
    int it = blockIdx.x & 15;   // 0..15
    int wave = threadIdx.x >> 5, lane = threadIdx.x & 31;
    int half = lane >> 4, lr = lane & 15;
    int koff = half * 2;

    int arow = it * 16 + lr;
    const float* ap = zs + arow * (STEP_ * DZ_) + s * DZ_ + koff;

    for (int t = 0; t < 2; ++t) {
        int jt = wave * 2 + t;
        int bcol = jt * 16 + lr;
        const float* lvp = zlv + bcol * (STEP_ * DZ_) + s * DZ_ + koff;
        const float* mp  = zm  + bcol * (STEP_ * DZ_) + s * DZ_ + koff;
        v8f acc = {0.f, 0.f, 0.f, 0.f, 0.f, 0.f, 0.f, 0.f};
        for (int k = 0; k < DZ_; k += 4) {
            float2 sa = *reinterpret_cast<const float2*>(ap + k);
            float2 lv = *reinterpret_cast<const float2*>(lvp + k);
            float2 mu = *reinterpret_cast<const float2*>(mp + k);
            v2f a1; a1.x = sa.x * sa.x;          a1.y = sa.y * sa.y;
            v2f b1; b1.x = __expf(-2.f * lv.x);  b1.y = __expf(-2.f * lv.y);
            v2f a2; a2.x = sa.x;                 a2.y = sa.y;
            v2f b2; b2.x = -2.f * mu.x * b1.x;   b2.y = -2.f * mu.y * b1.y;
            acc = wmma_f32(a1, b1, acc);
            acc = wmma_f32(a2, b2, acc);
        }
        float cj = cz[bcol * STEP_ + s];
        #pragma unroll
        for (int v = 0; v < 8; ++v) {
            int r = v + half * 8;
            float val = -0.5f * (acc[v] + cj);
            lz[r][bcol]  = val;
            lfz[r][bcol] = val + sumf[(it * 16 + r) * B_ + bcol];
        }
    }
    __syncthreads();

    // 16 threads per row i; each scans 16 of the 256 j's, then shuffle-reduce.
    int r = threadIdx.x >> 4, c = threadIdx.x & 15;
    float mz = -3.4e38f, mf = -3.4e38f;
    #pragma unroll
    for (int e = 0; e < 16; ++e) {
        mz = fmaxf(mz, lz[r][c * 16 + e]);
        mf = fmaxf(mf, lfz[r][c * 16 + e]);
    }
    for (int m = 8; m >= 1; m >>= 1) {
        mz = fmaxf(mz, __shfl_xor(mz, m, 32));
        mf = fmaxf(mf, __shfl_xor(mf, m, 32));
    }
    float sz = 0.f, sf = 0.f;
    #pragma unroll
    for (int e = 0; e < 16; ++e) {
        sz += __expf(lz[r][c * 16 + e] - mz);
        sf += __expf(lfz[r][c * 16 + e] - mf);
    }
    for (int m = 8; m >= 1; m >>= 1) {
        sz += __shfl_xor(sz, m, 32);
        sf += __shfl_xor(sf, m, 32);
    }
    if (c == 0) {
        int i = it * 16 + r;
        tz[s * B_ + i]  = fmaxf(-(mz + __logf(sz) - LOGC), 0.f);
        tfz[s * B_ + i] = fmaxf(-(mf + __logf(sf) - LOGC), 0.f);
    }
}

// ---------- final scalar: Hf + Hz - Hfz ----------
static __device__ float block_red(float v, float* red) {
    int t = threadIdx.x;
    red[t] = v;
    __syncthreads();
    for (int s = 128; s > 0; s >>= 1) {
        if (t < s) red[t] += red[t + s];
        __syncthreads();
    }
    float r = red[0];
    __syncthreads();
    return r;
}

__global__ void __launch_bounds__(256) k_final(const float* __restrict__ tf,
                                               const float* __restrict__ tz,
                                               const float* __restrict__ tfz,
                                               float* __restrict__ out) {
    __shared__ float red[256];
    float a = 0.f;
    for (int i = threadIdx.x; i < B_; i += 256) a += tf[i];
    float b = 0.f, c = 0.f;
    for (int i = threadIdx.x; i < STEP_ * B_; i += 256) { b += tz[i]; c += tfz[i]; }
    float Hf  = block_red(a, red) / (float)B_;
    float Hz  = block_red(b, red) / (float)(STEP_ * B_);
    float Hfz = block_red(c, red) / (float)(STEP_ * B_);
    if (threadIdx.x == 0) out[0] = Hf + Hz - Hfz;
}

extern "C" void kernel_launch(void* const* d_in, const int* in_sizes, int n_in,
                              void* d_out, int out_size, void* d_ws, size_t ws_size,
                              hipStream_t stream) {
    const float* fm  = (const float*)d_in[0];
    const float* flv = (const float*)d_in[1];
    const float* fs  = (const float*)d_in[2];
    const float* zm  = (const float*)d_in[3];
    const float* zlv = (const float*)d_in[4];
    const float* zs  = (const float*)d_in[5];

    float* W    = (float*)d_ws;
    float* cf   = W;                 // 256
    float* cz   = cf + 256;          // 8192
    float* sumf = cz + 8192;         // 65536
    float* tf   = sumf + 65536;      // 256
    float* tz   = tf + 256;          // 8192
    float* tfz  = tz + 8192;         // 8192   total ~362 KB

    k_const_f<<<1, 256, 0, stream>>>(fm, flv, cf);
    k_const_z<<<(B_ * STEP_) / 256, 256, 0, stream>>>(zm, zlv, cz);
    k_sumf<<<256, 32, 0, stream>>>(fm, flv, fs, cf, sumf);
    k_logqf<<<B_ / 8, 256, 0, stream>>>(sumf, tf);
    k_z<<<STEP_ * 16, 256, 0, stream>>>(zm, zlv, zs, cz, sumf, tz, tfz);
    k_final<<<1, 256, 0, stream>>>(tf, tz, tfz, (float*)d_out);
}